// WIoUBCELoss_66838281060455
// MI455X (gfx1250) — compile-verified
//
#include <hip/hip_runtime.h>
#include <math.h>

typedef float v2f __attribute__((ext_vector_type(2)));
typedef float v8f __attribute__((ext_vector_type(8)));

#define IMG     512
#define TILE    128
#define HALO_R  160   // 158 live rows (tile + 2*15) + 2 zero rows so K4-tiles never read garbage
#define HALO_P  162   // halo pitch: conflict-free row-per-lane horizontal pass
#define KP      162   // hsumT pitch along K: b64 B-loads conflict-free across the wave
#define NBLK    1024  // 64 images x 16 tiles

#define HALO_FLOATS  (HALO_R*HALO_P)
#define HSUMT_FLOATS (TILE*KP)
#define SMEM_FLOATS  (HALO_FLOATS + HSUMT_FLOATS + 32)

__global__ __launch_bounds__(256) void wiou_fused(
    const float* __restrict__ logits,
    const float* __restrict__ targets,
    float* __restrict__ partials)
{
    extern __shared__ float smem[];
    float* halo  = smem;                         // [HALO_R][HALO_P], row-major targets halo
    float* hsumT = smem + HALO_FLOATS;           // [TILE][KP], K-major horizontal sums
    float* red   = hsumT + HSUMT_FLOATS;

    const int img  = blockIdx.x >> 4;
    const int tile = blockIdx.x & 15;
    const int ty0  = (tile >> 2) * TILE;
    const int tx0  = (tile &  3) * TILE;
    const float* timg = targets + (size_t)img * IMG * IMG;
    const float* ximg = logits  + (size_t)img * IMG * IMG;

    // ---- Stage 1: stage zero-padded target halo into LDS.
    // Live cells go through GLOBAL_LOAD_ASYNC_TO_LDS_B32 (memory->LDS, no VGPR
    // round-trip, ASYNCcnt-tracked); the padding ring is zeroed with DS stores.
    // The two sets of cells are disjoint, so no async/DS ordering hazard.
    const unsigned lds_base = __builtin_amdgcn_groupstaticsize();
    for (int i = threadIdx.x; i < HALO_R*HALO_P; i += 256) {
        const int hy = i / HALO_P, hx = i - hy*HALO_P;
        const int gy = ty0 + hy - 15, gx = tx0 + hx - 15;
        const bool live = (hy < 158) && (hx < 158) &&
                          ((unsigned)gy < IMG) && ((unsigned)gx < IMG);
        if (live) {
            const unsigned lds_off = lds_base + (unsigned)i * 4u;
            const unsigned goff    = (unsigned)(gy*IMG + gx) * 4u;
            asm volatile("global_load_async_to_lds_b32 %0, %1, %2"
                         :: "v"(lds_off), "v"(goff), "s"(timg) : "memory");
        } else {
            halo[i] = 0.f;
        }
    }
    asm volatile("s_wait_asynccnt 0" ::: "memory");
    __syncthreads();

    // ---- Stage 2: horizontal 31-wide running-window sums, written K-major.
    // Per x-step the 32 lanes (consecutive y) write consecutive LDS words.
    for (int y = threadIdx.x; y < HALO_R; y += 256) {
        const float* row = halo + y*HALO_P;
        float s = 0.f;
        #pragma unroll
        for (int i = 0; i < 30; ++i) s += row[i];
        for (int x = 0; x < TILE; ++x) {
            s += row[x + 30];
            hsumT[x*KP + y] = s;
            s -= row[x];
        }
    }
    __syncthreads();

    // ---- Stage 3: vertical 31-sum as banded-ones matmul via V_WMMA_F32_16X16X4_F32,
    // fused with sigmoid/BCE/IoU elementwise math and partial reduction.
    const int wave  = threadIdx.x >> 5;
    const int lane  = threadIdx.x & 31;
    const int lh    = lane & 15;
    const int khi   = (lane < 16) ? 0 : 2;          // A/B: upper half-wave holds K2,K3
    const int m0    = wave * 16;                    // each wave owns one 16-row strip
    const int mbase = m0 + ((lane < 16) ? 0 : 8);   // C/D: upper half-wave holds M+8

    // A-band tiles: A[m][k] = 1 iff m <= k <= m+30; lane-constant (rel = 4kk+khi-lh).
    v2f a[12];
    #pragma unroll
    for (int kk = 0; kk < 12; ++kk) {
        const int rel = 4*kk + khi - lh;
        a[kk].x = (rel     >= 0 && rel     <= 30) ? 1.f : 0.f;
        a[kk].y = (rel + 1 >= 0 && rel + 1 <= 30) ? 1.f : 0.f;
    }

    float accW = 0.f, accI = 0.f, accT = 0.f;

    for (int nt = 0; nt < 8; ++nt) {
        const int n = nt*16 + lh;                   // this lane's N column

        // Prefetch this lane's 8-row logits column FIRST: the HBM latency is
        // then covered by the 6 ds_load_2addr_b64 + 12-WMMA chain below.
        float xv[8];
        const float* xcol = ximg + (ty0 + mbase)*IMG + (tx0 + n);
        #pragma unroll
        for (int r = 0; r < 8; ++r) xv[r] = xcol[r*IMG];

        // Batch all 12 B pairs as b64 loads (8B-aligned: n*KP+m0+4kk+khi is even),
        // then run the 12-WMMA accumulation chain.
        const float* bcol = hsumT + n*KP + m0 + khi;
        v2f bb[12];
        #pragma unroll
        for (int kk = 0; kk < 12; ++kk)
            bb[kk] = *(const v2f*)(bcol + 4*kk);
        v8f c = {};
        #pragma unroll
        for (int kk = 0; kk < 12; ++kk)
            c = __builtin_amdgcn_wmma_f32_16x16x4_f32(
                    false, a[kk], false, bb[kk], (short)0, c, false, false);

        // Batch the 8 target reads from the LDS halo.
        float tv[8];
        #pragma unroll
        for (int r = 0; r < 8; ++r)
            tv[r] = halo[(mbase + r + 15)*HALO_P + (n + 15)];

        #pragma unroll
        for (int r = 0; r < 8; ++r) {
            const float box = c[r] * (1.0f/961.0f);              // avg_pool value
            const float t   = tv[r];
            const float w   = 1.f + 5.f*fabsf(box - t);          // boundary weight
            const float x   = xv[r];
            // Hardware-transcendental BCE chain: v_exp_f32 / v_log_f32.
            // log(1+e) loses log1p protection only when e < ~1e-7 (|x| > 16),
            // where the term itself is < 1e-7 -- negligible in absolute error.
            const float e   = __expf(-fabsf(x));
            const float l1p = __logf(1.f + e);
            const float log_p   = fminf(x, 0.f)  - l1p;          // log sigmoid(x)
            const float log_1mp = fminf(-x, 0.f) - l1p;          // log sigmoid(-x)
            const float bce = -(t*log_p + (1.f - t)*log_1mp);
            const float p   = __expf(log_p);                     // sigmoid(x)
            accW = fmaf(w, bce,   accW);
            accI = fmaf(p*t, w,   accI);
            accT = fmaf(p + t, w, accT);
        }
    }

    // ---- deterministic block reduction: wave shuffles -> LDS -> one writer ----
    #pragma unroll
    for (int off = 16; off > 0; off >>= 1) {
        accW += __shfl_down(accW, off, 32);
        accI += __shfl_down(accI, off, 32);
        accT += __shfl_down(accT, off, 32);
    }
    if (lane == 0) { red[wave] = accW; red[8+wave] = accI; red[16+wave] = accT; }
    __syncthreads();
    if (threadIdx.x == 0) {
        float sW = 0.f, sI = 0.f, sT = 0.f;
        #pragma unroll
        for (int w = 0; w < 8; ++w) { sW += red[w]; sI += red[8+w]; sT += red[16+w]; }
        partials[          blockIdx.x] = sW;
        partials[  NBLK + blockIdx.x] = sI;
        partials[2*NBLK + blockIdx.x] = sT;
    }
}

__global__ __launch_bounds__(256) void wiou_finalize(
    const float* __restrict__ partials, float* __restrict__ out)
{
    __shared__ float red[24];
    float sW = 0.f, sI = 0.f, sT = 0.f;
    for (int i = threadIdx.x; i < NBLK; i += 256) {
        sW += partials[i];
        sI += partials[NBLK + i];
        sT += partials[2*NBLK + i];
    }
    const int wave = threadIdx.x >> 5, lane = threadIdx.x & 31;
    #pragma unroll
    for (int off = 16; off > 0; off >>= 1) {
        sW += __shfl_down(sW, off, 32);
        sI += __shfl_down(sI, off, 32);
        sT += __shfl_down(sT, off, 32);
    }
    if (lane == 0) { red[wave] = sW; red[8+wave] = sI; red[16+wave] = sT; }
    __syncthreads();
    if (threadIdx.x == 0) {
        float W = 0.f, I = 0.f, T = 0.f;
        #pragma unroll
        for (int w = 0; w < 8; ++w) { W += red[w]; I += red[8+w]; T += red[16+w]; }
        const float nelem = 64.0f * 512.0f * 512.0f;
        const float wbce  = W / nelem;
        const float uni   = T - I;
        const float wiou  = 1.f - (I + 1.f) / (uni + 1.f);
        out[0] = wbce + wiou;
    }
}

extern "C" void kernel_launch(void* const* d_in, const int* in_sizes, int n_in,
                              void* d_out, int out_size, void* d_ws, size_t ws_size,
                              hipStream_t stream) {
    (void)in_sizes; (void)n_in; (void)out_size; (void)ws_size;
    const float* logits  = (const float*)d_in[0];   // "inputs"
    const float* targets = (const float*)d_in[1];   // "targets"
    float* partials = (float*)d_ws;                  // 3 * 1024 floats = 12 KB
    float* out      = (float*)d_out;

    const size_t smem = SMEM_FLOATS * sizeof(float); // ~187 KB, fits 320 KB/WGP
    (void)hipFuncSetAttribute((const void*)wiou_fused,
                              hipFuncAttributeMaxDynamicSharedMemorySize, (int)smem);

    wiou_fused<<<dim3(NBLK), dim3(256), smem, stream>>>(logits, targets, partials);
    wiou_finalize<<<dim3(1), dim3(256), 0, stream>>>(partials, out);
}